// ModelNew_1580547968669
// MI455X (gfx1250) — compile-verified
//
#include <hip/hip_runtime.h>

typedef __attribute__((ext_vector_type(2))) float v2f;
typedef __attribute__((ext_vector_type(8))) float v8f;

#define WAVES_PER_BLOCK 8
#define XT_STRIDE 34                    // padded [sp][ci] stride (floats)
#define XT_FLOATS (64 * XT_STRIDE)      // 2176 floats per wave tile
#define WA_FLOATS (27 * 8 * 32 * 2)     // 13824 floats of rearranged weights

__global__ __launch_bounds__(256)
void convt3d_pool_sum_wmma(const float* __restrict__ x,
                           const float* __restrict__ W,
                           const float* __restrict__ bias,
                           float* __restrict__ out)
{
    __shared__ __align__(16) float sWA[WA_FLOATS];
    __shared__ __align__(16) float sX[WAVES_PER_BLOCK * XT_FLOATS];

    const int tid  = threadIdx.x;
    const int lane = tid & 31;
    const int wave = tid >> 5;

    // ---- rearrange W (32ci,16co,27tap) into A-fragment layout in LDS ----
    // sWA[((tap*8+chunk)*32 + lane)*2 + j] = W[ci=chunk*4+2*(lane>>4)+j][co=lane&15][tap]
    for (int idx = tid; idx < WA_FLOATS; idx += 256) {
        int j     = idx & 1;
        int l     = (idx >> 1) & 31;
        int chunk = (idx >> 6) & 7;
        int tap   = idx >> 9;
        int ci    = chunk * 4 + ((l >> 4) << 1) + j;
        int co    = l & 15;
        sWA[idx] = W[ci * 432 + co * 27 + tap];
    }

    // ---- this wave's pool cell ----
    int cell = blockIdx.x * WAVES_PER_BLOCK + wave;   // 0..7999
    int n  = cell / 1000;
    int r3 = cell - n * 1000;
    int pd = r3 / 100;
    int r2 = r3 - pd * 100;
    int ph = r2 / 10;
    int pw = r2 - ph * 10;

    // ---- load x tile: x[n, 0..31, 3pd..3pd+3, 3ph..+3, 3pw..+3] -> LDS [sp][ci] ----
    float* xt = sX + wave * XT_FLOATS;
    const float* xb = x + n * 1048576 + (3 * pd) * 1024 + (3 * ph) * 32 + (3 * pw);
    for (int it = 0; it < 64; ++it) {
        int flat = it * 32 + lane;          // ci-major flattening, 0..2047
        int ci = flat >> 6;
        int sp = flat & 63;
        int a  = sp >> 4;
        int bh = (sp >> 2) & 3;
        int c  = sp & 3;
        xt[sp * XT_STRIDE + ci] = xb[ci * 32768 + a * 1024 + bh * 32 + c];
    }
    __syncthreads();

    // ---- precompute per-tile packed (d_off+1, h_off+1, w_off+1); -1 = pad lane ----
    int pk[14];
#pragma unroll
    for (int t = 0; t < 14; ++t) {
        int p  = t * 16 + (lane & 15);      // conv-output position within 6x6x6 block
        int d1 = p / 36;
        int rr = p - d1 * 36;
        int h1 = rr / 6;
        int w1 = rr - h1 * 6;
        int packed = (d1 + 1) | ((h1 + 1) << 8) | ((w1 + 1) << 16);
        pk[t] = (p < 216) ? packed : -1;
    }

    const int half2 = (lane >> 4) << 1;     // ci sub-offset inside a K=4 chunk: 0 or 2

    v8f acc[14];
#pragma unroll
    for (int t = 0; t < 14; ++t)
        acc[t] = (v8f){0.f, 0.f, 0.f, 0.f, 0.f, 0.f, 0.f, 0.f};

    for (int tap = 0; tap < 27; ++tap) {
        int kd = tap / 9;
        int rk = tap - kd * 9;
        int kh = rk / 3;
        int kw = rk - kh * 3;

        // A fragments for the 8 ci-chunks of this tap
        v2f A[8];
#pragma unroll
        for (int c = 0; c < 8; ++c)
            A[c] = *(const v2f*)(sWA + tap * 512 + c * 64 + lane * 2);

#pragma unroll
        for (int t = 0; t < 14; ++t) {
            int pkt = pk[t];
            int ed = (pkt & 255)         - kd;    // e = off+1-k; valid iff e>=0 and even
            int eh = ((pkt >> 8) & 255)  - kh;
            int ew = ((pkt >> 16) & 255) - kw;
            int m  = ed | eh | ew;
            bool valid = (((m | pkt) >= 0) && ((m & 1) == 0));
            int sp  = ((ed >> 1) << 4) + ((eh >> 1) << 2) + (ew >> 1);
            int off = valid ? (sp * XT_STRIDE + half2) : half2;
#pragma unroll
            for (int c = 0; c < 8; ++c) {
                v2f Br = *(const v2f*)(xt + off + c * 4);
                v2f B;
                B.x = valid ? Br.x : 0.0f;
                B.y = valid ? Br.y : 0.0f;
                acc[t] = __builtin_amdgcn_wmma_f32_16x16x4_f32(
                    false, A[c], false, B, (short)0, acc[t], false, false);
            }
        }
    }

    // ---- maxpool over 216 positions, + bias, sum over 16 co ----
    const float NEG = -3.402823466e38f;
    float s = 0.0f;
    int hof = (lane >> 4) << 3;             // co offset: 0 or 8
#pragma unroll
    for (int r = 0; r < 8; ++r) {
        float mx = NEG;
#pragma unroll
        for (int t = 0; t < 13; ++t)
            mx = fmaxf(mx, acc[t][r]);
        float v13 = ((lane & 15) < 8) ? acc[13][r] : NEG;   // tile 13: only 8 valid positions
        mx = fmaxf(mx, v13);
        mx = fmaxf(mx, __shfl_xor(mx, 1, 32));
        mx = fmaxf(mx, __shfl_xor(mx, 2, 32));
        mx = fmaxf(mx, __shfl_xor(mx, 4, 32));
        mx = fmaxf(mx, __shfl_xor(mx, 8, 32));
        s += mx + bias[r + hof];
    }
    s += __shfl_xor(s, 16, 32);             // combine co 0-7 (lanes<16) with co 8-15
    if (lane == 0) out[cell] = s;
}

extern "C" void kernel_launch(void* const* d_in, const int* in_sizes, int n_in,
                              void* d_out, int out_size, void* d_ws, size_t ws_size,
                              hipStream_t stream) {
    const float* x = (const float*)d_in[0];   // (8,32,32,32,32) f32
    const float* W = (const float*)d_in[1];   // (32,16,3,3,3)  f32
    const float* b = (const float*)d_in[2];   // (16,)          f32
    float* out = (float*)d_out;               // 8000 f32

    dim3 grid(1000), block(256);
    hipLaunchKernelGGL(convt3d_pool_sum_wmma, grid, block, 0, stream, x, W, b, out);
}